// KGCN_33526514712971
// MI455X (gfx1250) — compile-verified
//
#include <hip/hip_runtime.h>
#include <hip/hip_bf16.h>
#include <math.h>

#define N_NODES 50000
#define N_EDGES 800000
#define D       64
#define D_OUT   64
#define K_TOT   (2 * D)          // 128

typedef __attribute__((ext_vector_type(2))) float v2f;
typedef __attribute__((ext_vector_type(8))) float v8f;

// ---------------------------------------------------------------------------
// Kernel 1: per-edge score = dot(h_src[src], e), exp, segment-sum into denom.
// One wave32 per edge: lane loads 2 of the 64 features (fully coalesced 256B
// row reads), butterfly reduction, lane 0 does the exp + atomic.
// h_src (12.8MB) stays L2-resident; e (205MB) streams once.
// ---------------------------------------------------------------------------
__global__ void __launch_bounds__(256) edge_score_kernel(
    const float* __restrict__ h_src, const float* __restrict__ e,
    const int* __restrict__ src, const int* __restrict__ dst,
    float* __restrict__ exp_s, float* __restrict__ denom)
{
    const int edge = (int)((blockIdx.x * 256u + threadIdx.x) >> 5);
    const int lane = threadIdx.x & 31;
    if (edge >= N_EDGES) return;

    const int s = src[edge];
    const float* __restrict__ hu = h_src + (size_t)s    * D;
    const float* __restrict__ ef = e     + (size_t)edge * D;

    float p = hu[lane] * ef[lane] + hu[lane + 32] * ef[lane + 32];
#pragma unroll
    for (int off = 16; off > 0; off >>= 1)
        p += __shfl_xor(p, off, 32);

    if (lane == 0) {
        const float ex = expf(p);            // raw exp, as in reference
        exp_s[edge] = ex;
        atomicAdd(&denom[dst[edge]], ex);
    }
}

// ---------------------------------------------------------------------------
// Kernel 2: pi = exp_s / denom[dst]; scatter-add pi * h_src[src] into h_sum.
// One wave32 per edge again; 64 f32 atomics/edge land in L2 (h_sum = 12.8MB).
// ---------------------------------------------------------------------------
__global__ void __launch_bounds__(256) edge_aggregate_kernel(
    const float* __restrict__ h_src, const int* __restrict__ src,
    const int* __restrict__ dst, const float* __restrict__ exp_s,
    const float* __restrict__ denom, float* __restrict__ h_sum)
{
    const int edge = (int)((blockIdx.x * 256u + threadIdx.x) >> 5);
    const int lane = threadIdx.x & 31;
    if (edge >= N_EDGES) return;

    const int s = src[edge];
    const int d = dst[edge];
    const float pi = exp_s[edge] / denom[d];

    const float* __restrict__ hu = h_src + (size_t)s * D;
    float* __restrict__ hs = h_sum + (size_t)d * D;

    atomicAdd(&hs[lane],      hu[lane]      * pi);
    atomicAdd(&hs[lane + 32], hu[lane + 32] * pi);
}

// ---------------------------------------------------------------------------
// Kernel 3: out = [h_dst | h_sum] @ W^T + b  via V_WMMA_F32_16X16X4_F32.
// Block = 128 threads (4 waves). W[64][128] staged in LDS with a 132-dword
// row stride -> B-fragment loads (lane n reads W[n][k..k+1]) hit distinct
// banks. Each wave computes a 16-row x 64-col tile: 4 accumulators, K=128
// in steps of 4 => 128 WMMAs per wave.
//
// f32 16x16x4 operand layout (ISA 7.12.2):
//   A (16x4): lane m=lane&15 is row M; VGPR0/1 hold K = k0+2*hi, k0+2*hi+1
//   B (4x16): lane n=lane&15 is col N; VGPR0/1 hold rows K = k0+2*hi, +1
//   C/D     : VGPR r -> (M = r + 8*hi, N = lane&15)
// ---------------------------------------------------------------------------
#define LDSW_STRIDE 132   // 128 + 4 dwords pad: conflict-free, 8B-aligned rows

__global__ void __launch_bounds__(128) gat_linear_wmma_kernel(
    const float* __restrict__ h_dst, const float* __restrict__ h_sum,
    const float* __restrict__ W, const float* __restrict__ bias,
    float* __restrict__ out)
{
    __shared__ float ldsW[D_OUT * LDSW_STRIDE];

    const int tid = threadIdx.x;
    // Cooperative, coalesced stage of W [D_OUT x 128] into padded LDS.
    for (int idx = tid; idx < D_OUT * K_TOT; idx += 128) {
        const int col = idx >> 7;      // output channel (row of W)
        const int k   = idx & 127;
        ldsW[col * LDSW_STRIDE + k] = W[idx];
    }
    __syncthreads();

    const int lane = tid & 31;
    const int wave = tid >> 5;
    const int m    = lane & 15;        // M (A rows) and N (B cols) index
    const int hi   = lane >> 4;        // half-wave select (K pair / M+8)

    const int rowBase = (blockIdx.x * 4 + wave) * 16;
    if (rowBase >= N_NODES) return;    // wave-uniform: EXEC stays all-ones

    const float* __restrict__ arow0 = h_dst + (size_t)(rowBase + m) * D;
    const float* __restrict__ arow1 = h_sum + (size_t)(rowBase + m) * D;

    v8f acc[4] = {};

    // K = 0..63 from h_dst
#pragma unroll
    for (int k0 = 0; k0 < 64; k0 += 4) {
        const int ka = k0 + hi * 2;
        const v2f a = *(const v2f*)(arow0 + ka);
#pragma unroll
        for (int nt = 0; nt < 4; ++nt) {
            const v2f bf = *(const v2f*)(&ldsW[(nt * 16 + m) * LDSW_STRIDE + ka]);
            acc[nt] = __builtin_amdgcn_wmma_f32_16x16x4_f32(
                false, a, false, bf, (short)0, acc[nt], false, false);
        }
    }
    // K = 64..127 from h_sum
#pragma unroll
    for (int k0 = 0; k0 < 64; k0 += 4) {
        const int ka = k0 + hi * 2;
        const v2f a = *(const v2f*)(arow1 + ka);
#pragma unroll
        for (int nt = 0; nt < 4; ++nt) {
            const v2f bf = *(const v2f*)(&ldsW[(nt * 16 + m) * LDSW_STRIDE + 64 + ka]);
            acc[nt] = __builtin_amdgcn_wmma_f32_16x16x4_f32(
                false, a, false, bf, (short)0, acc[nt], false, false);
        }
    }

    // Bias + store per the C/D VGPR layout.
#pragma unroll
    for (int nt = 0; nt < 4; ++nt) {
        const float bv = bias[nt * 16 + m];
#pragma unroll
        for (int r = 0; r < 8; ++r) {
            const int orow = rowBase + r + hi * 8;
            out[(size_t)orow * D_OUT + nt * 16 + m] = acc[nt][r] + bv;
        }
    }
}

// ---------------------------------------------------------------------------
extern "C" void kernel_launch(void* const* d_in, const int* in_sizes, int n_in,
                              void* d_out, int out_size, void* d_ws, size_t ws_size,
                              hipStream_t stream)
{
    const float* h_src = (const float*)d_in[0];
    const float* h_dst = (const float*)d_in[1];
    const float* e     = (const float*)d_in[2];
    const int*   src   = (const int*)  d_in[3];
    const int*   dst   = (const int*)  d_in[4];
    const float* W     = (const float*)d_in[5];
    const float* bias  = (const float*)d_in[6];
    float* out = (float*)d_out;

    // Workspace: denom[N] | exp_s[E] | h_sum[N*D]  (~16.2 MB)
    float* denom = (float*)d_ws;
    float* exp_s = denom + N_NODES;
    float* h_sum = exp_s + N_EDGES;

    // Accumulation buffers must be zeroed every call (graph-capture safe).
    hipMemsetAsync(denom, 0, (size_t)N_NODES * sizeof(float), stream);
    hipMemsetAsync(h_sum, 0, (size_t)N_NODES * D * sizeof(float), stream);

    const int edgeBlocks = (N_EDGES + 7) / 8;   // 8 edges (waves) per 256-thr block
    edge_score_kernel<<<edgeBlocks, 256, 0, stream>>>(h_src, e, src, dst, exp_s, denom);
    edge_aggregate_kernel<<<edgeBlocks, 256, 0, stream>>>(h_src, src, dst, exp_s, denom, h_sum);

    const int gemmBlocks = (N_NODES + 63) / 64; // 4 waves x 16 rows per block
    gat_linear_wmma_kernel<<<gemmBlocks, 128, 0, stream>>>(h_dst, h_sum, W, bias, out);
}